// GAT2_26465588478554
// MI455X (gfx1250) — compile-verified
//
#include <hip/hip_runtime.h>
#include <hip/hip_bf16.h>

// GATv2 x2 + BN + heads, fp32 throughout.
// GEMMs use native CDNA5 fp32 WMMA (v_wmma_f32_16x16x4_f32).
// Edge phases are wave-per-edge with float4-coalesced row gathers (L2-resident).

typedef __attribute__((ext_vector_type(2))) float v2f;
typedef __attribute__((ext_vector_type(8))) float v8f;

#define NEG_SLOPE 0.2f
#define BN_EPS 1e-5f

// ---------- monotone float<->uint encoding for atomicMax on floats ----------
__device__ inline unsigned enc_f(float f) {
    unsigned b = __float_as_uint(f);
    return (b & 0x80000000u) ? ~b : (b | 0x80000000u);
}
__device__ inline float dec_f(unsigned u) {
    return __uint_as_float((u & 0x80000000u) ? (u & 0x7FFFFFFFu) : ~u);
}

// ---------------------------- fill kernels ----------------------------------
__global__ void fill_f32(float* __restrict__ p, float v, long long n) {
    long long i = (long long)blockIdx.x * blockDim.x + threadIdx.x;
    if (i < n) p[i] = v;
}
__global__ void fill_u32(unsigned* __restrict__ p, unsigned v, long long n) {
    long long i = (long long)blockIdx.x * blockDim.x + threadIdx.x;
    if (i < n) p[i] = v;
}
__global__ void add_vec(const float* __restrict__ a, const float* __restrict__ b,
                        float* __restrict__ o, int n) {
    int i = blockIdx.x * blockDim.x + threadIdx.x;
    if (i < n) o[i] = a[i] + b[i];
}

// ---------------- self-loop mean edge-attr (PyG fill_value='mean') ----------
__global__ void edge_stats(const int* __restrict__ ei, const float* __restrict__ eattr,
                           float* __restrict__ eaf, float* __restrict__ deg,
                           float* __restrict__ asum, int E0) {
    int e = blockIdx.x * blockDim.x + threadIdx.x;
    if (e >= E0) return;
    float v = eattr[e];
    eaf[e] = v;
    int d = ei[E0 + e];
    atomicAdd(deg + d, 1.0f);
    atomicAdd(asum + d, v);
}
__global__ void mean_loop(const float* __restrict__ deg, const float* __restrict__ asum,
                          float* __restrict__ eaf_loop, int N) {
    int i = blockIdx.x * blockDim.x + threadIdx.x;
    if (i < N) eaf_loop[i] = asum[i] / fmaxf(deg[i], 1.0f);
}

// ------------------------- fp32 WMMA GEMM -----------------------------------
// out[M x Nc] = A[M x K] @ W[K x Nc] (+bias) (+relu). M%16==0, K%4==0, Nc%16==0.
// One wave computes one 16x16 tile via v_wmma_f32_16x16x4_f32.
// A 16x4 layout: lane(0-15) row=lane, v0=K0,v1=K1; lane(16-31) v0=K2,v1=K3.
// B 4x16 layout: lane(0-15) col=lane, v0=K0,v1=K1; lane(16-31) v0=K2,v1=K3.
// C/D 16x16: VGPR r -> row = r + 8*(lane>=16), col = lane&15.
__global__ void gemm_wmma_f32(const float* __restrict__ A, const float* __restrict__ W,
                              const float* __restrict__ bias, float* __restrict__ out,
                              int M, int K, int Nc, int act) {
    int wave = (blockIdx.x * blockDim.x + threadIdx.x) >> 5;
    int colTiles = Nc >> 4;
    int totalTiles = (M >> 4) * colTiles;
    if (wave >= totalTiles) return;
    int rowT = wave / colTiles;
    int colT = wave - rowT * colTiles;
    int row0 = rowT << 4, col0 = colT << 4;
    int lane = threadIdx.x & 31;
    int half = lane >> 4;        // 0: K pair {0,1}; 1: K pair {2,3}
    int lrow = lane & 15;

    v8f acc = {};
    const float* arow = A + (size_t)(row0 + lrow) * K;
    for (int kk = 0; kk < K; kk += 4) {
        int ka = kk + 2 * half;
        v2f a, b;
        a.x = arow[ka];
        a.y = arow[ka + 1];
        const float* bp = W + (size_t)ka * Nc + col0 + lrow;
        b.x = bp[0];
        b.y = bp[Nc];
        acc = __builtin_amdgcn_wmma_f32_16x16x4_f32(
            false, a, false, b, (short)0, acc, false, false);
    }
    float bv = bias ? bias[col0 + lrow] : 0.0f;
#pragma unroll
    for (int r = 0; r < 8; ++r) {
        int row = row0 + r + half * 8;
        float v = acc[r] + bv;
        if (act == 1) v = fmaxf(v, 0.0f);
        out[(size_t)row * Nc + col0 + lrow] = v;
    }
}

// --------------------- GAT layer 1 (H=4, C=32, HC=128) ----------------------
// wave per edge; lane l covers channels [4l, 4l+3]; head = l>>3
__global__ void gat1_edge_alpha(const int* __restrict__ ei, const float* __restrict__ eaf,
                                const float* __restrict__ xl, const float* __restrict__ xr,
                                const float* __restrict__ We, const float* __restrict__ att,
                                float* __restrict__ alpha, unsigned* __restrict__ amax,
                                int E0, int Etot) {
    int e = (blockIdx.x * blockDim.x + threadIdx.x) >> 5;
    if (e >= Etot) return;
    int lane = threadIdx.x & 31;
    int s, d;
    if (e < E0) { s = ei[e]; d = ei[E0 + e]; } else { s = d = e - E0; }
    float eav = eaf[e];
    float4 a4 = ((const float4*)(xl + (size_t)s * 128))[lane];
    float4 b4 = ((const float4*)(xr + (size_t)d * 128))[lane];
    float4 w4 = ((const float4*)We)[lane];
    float4 t4 = ((const float4*)att)[lane];
    float p = 0.0f, m;
    m = a4.x + b4.x + eav * w4.x; m = (m > 0.0f) ? m : NEG_SLOPE * m; p += m * t4.x;
    m = a4.y + b4.y + eav * w4.y; m = (m > 0.0f) ? m : NEG_SLOPE * m; p += m * t4.y;
    m = a4.z + b4.z + eav * w4.z; m = (m > 0.0f) ? m : NEG_SLOPE * m; p += m * t4.z;
    m = a4.w + b4.w + eav * w4.w; m = (m > 0.0f) ? m : NEG_SLOPE * m; p += m * t4.w;
    p += __shfl_xor(p, 1, 32);
    p += __shfl_xor(p, 2, 32);
    p += __shfl_xor(p, 4, 32);   // per-head sum within aligned 8-lane groups
    if ((lane & 7) == 0) {
        int h = lane >> 3;
        alpha[(size_t)e * 4 + h] = p;
        atomicMax(amax + (size_t)d * 4 + h, enc_f(p));
    }
}

// shared for both layers: exp(alpha - amax[dst]) and segment-sum denominator
__global__ void softmax_num(const int* __restrict__ ei_dst, const unsigned* __restrict__ amax,
                            float* __restrict__ alpha, float* __restrict__ den,
                            int E0, int Etot, int H) {
    int idx = blockIdx.x * blockDim.x + threadIdx.x;
    if (idx >= Etot * H) return;
    int e = idx / H;
    int h = idx - e * H;
    int d = (e < E0) ? ei_dst[e] : (e - E0);
    float ex = expf(alpha[idx] - dec_f(amax[(size_t)d * H + h]));
    alpha[idx] = ex;
    atomicAdd(den + (size_t)d * H + h, ex);
}

__global__ void gat1_scatter(const int* __restrict__ ei, const float* __restrict__ xl,
                             const float* __restrict__ ex, const float* __restrict__ den,
                             float* __restrict__ hacc, int E0, int Etot) {
    int e = (blockIdx.x * blockDim.x + threadIdx.x) >> 5;
    if (e >= Etot) return;
    int lane = threadIdx.x & 31;
    int s, d;
    if (e < E0) { s = ei[e]; d = ei[E0 + e]; } else { s = d = e - E0; }
    int h = lane >> 3;
    float a = ex[(size_t)e * 4 + h] / (den[(size_t)d * 4 + h] + 1e-16f);
    float4 v = ((const float4*)(xl + (size_t)s * 128))[lane];
    float* o = hacc + (size_t)d * 128 + lane * 4;
    atomicAdd(o + 0, a * v.x);
    atomicAdd(o + 1, a * v.y);
    atomicAdd(o + 2, a * v.z);
    atomicAdd(o + 3, a * v.w);
}

// ------------------------------ BatchNorm -----------------------------------
__global__ void bn_stats(const float* __restrict__ h, float* __restrict__ stats, int N) {
    int c = threadIdx.x & 127;
    int r0 = blockIdx.x * 2 + (threadIdx.x >> 7);
    float s = 0.0f, s2 = 0.0f;
    for (int r = r0; r < N; r += gridDim.x * 2) {
        float v = h[(size_t)r * 128 + c];
        s += v; s2 += v * v;
    }
    atomicAdd(stats + c, s);
    atomicAdd(stats + 128 + c, s2);
}
__global__ void bn_apply_elu(float* __restrict__ h, const float* __restrict__ stats,
                             const float* __restrict__ g, const float* __restrict__ b,
                             float invN, long long total) {
    long long idx = (long long)blockIdx.x * blockDim.x + threadIdx.x;
    if (idx >= total) return;
    int c = (int)(idx & 127);
    float mu = stats[c] * invN;
    float var = stats[128 + c] * invN - mu * mu;
    float v = (h[idx] - mu) * rsqrtf(var + BN_EPS) * g[c] + b[c];
    h[idx] = (v > 0.0f) ? v : (expf(v) - 1.0f);
}

// --------------------- GAT layer 2 (H=1, C=32) ------------------------------
__global__ void gat2_edge_alpha(const int* __restrict__ ei, const float* __restrict__ eaf,
                                const float* __restrict__ xl, const float* __restrict__ xr,
                                const float* __restrict__ We, const float* __restrict__ att,
                                float* __restrict__ alpha, unsigned* __restrict__ amax,
                                int E0, int Etot) {
    int e = (blockIdx.x * blockDim.x + threadIdx.x) >> 5;
    if (e >= Etot) return;
    int lane = threadIdx.x & 31;
    int s, d;
    if (e < E0) { s = ei[e]; d = ei[E0 + e]; } else { s = d = e - E0; }
    float eav = eaf[e];
    float m = xl[(size_t)s * 32 + lane] + xr[(size_t)d * 32 + lane] + eav * We[lane];
    m = (m > 0.0f) ? m : NEG_SLOPE * m;
    float p = m * att[lane];
    p += __shfl_xor(p, 1, 32);
    p += __shfl_xor(p, 2, 32);
    p += __shfl_xor(p, 4, 32);
    p += __shfl_xor(p, 8, 32);
    p += __shfl_xor(p, 16, 32);
    if (lane == 0) {
        alpha[e] = p;
        atomicMax(amax + d, enc_f(p));
    }
}
__global__ void gat2_scatter(const int* __restrict__ ei, const float* __restrict__ xl,
                             const float* __restrict__ ex, const float* __restrict__ den,
                             float* __restrict__ out2, float* __restrict__ w_out,
                             int E0, int Etot) {
    int e = (blockIdx.x * blockDim.x + threadIdx.x) >> 5;
    if (e >= Etot) return;
    int lane = threadIdx.x & 31;
    int s, d;
    if (e < E0) { s = ei[e]; d = ei[E0 + e]; } else { s = d = e - E0; }
    float a = ex[e] / (den[d] + 1e-16f);
    if (lane == 0) w_out[e] = a;
    atomicAdd(out2 + (size_t)d * 32 + lane, a * xl[(size_t)s * 32 + lane]);
}

__global__ void bias_elu(float* __restrict__ h, const float* __restrict__ b,
                         long long total, int C) {
    long long idx = (long long)blockIdx.x * blockDim.x + threadIdx.x;
    if (idx >= total) return;
    int c = (int)(idx % C);
    float v = h[idx] + b[c];
    h[idx] = (v > 0.0f) ? v : (expf(v) - 1.0f);
}

// --------------------------- heads + log_softmax ----------------------------
__global__ void heads_kernel(const float* __restrict__ hl,
                             const float* __restrict__ Wc, const float* __restrict__ bc,
                             const float* __restrict__ Wt, const float* __restrict__ bt,
                             float* __restrict__ out, int N) {
    int i = blockIdx.x * blockDim.x + threadIdx.x;
    if (i >= N) return;
    float h[32];
#pragma unroll
    for (int c = 0; c < 32; ++c) h[c] = hl[(size_t)i * 32 + c];

    // clone head: 19 classes
    float lc[19];
    float mx = -1e30f;
#pragma unroll
    for (int j = 0; j < 19; ++j) {
        float s = bc[j];
#pragma unroll
        for (int c = 0; c < 32; ++c) s += h[c] * Wc[c * 19 + j];
        lc[j] = s;
        mx = fmaxf(mx, s);
    }
    float sum = 0.0f;
#pragma unroll
    for (int j = 0; j < 19; ++j) sum += expf(lc[j] - mx);
    float lse = mx + logf(sum);
#pragma unroll
    for (int j = 0; j < 19; ++j) out[(size_t)i * 48 + j] = lc[j] - lse;

    // type head: 29 classes
    float lt[29];
    mx = -1e30f;
#pragma unroll
    for (int j = 0; j < 29; ++j) {
        float s = bt[j];
#pragma unroll
        for (int c = 0; c < 32; ++c) s += h[c] * Wt[c * 29 + j];
        lt[j] = s;
        mx = fmaxf(mx, s);
    }
    sum = 0.0f;
#pragma unroll
    for (int j = 0; j < 29; ++j) sum += expf(lt[j] - mx);
    lse = mx + logf(sum);
#pragma unroll
    for (int j = 0; j < 29; ++j) out[(size_t)i * 48 + 19 + j] = lt[j] - lse;
}

// ----------------------------------------------------------------------------
extern "C" void kernel_launch(void* const* d_in, const int* in_sizes, int n_in,
                              void* d_out, int out_size, void* d_ws, size_t ws_size,
                              hipStream_t stream) {
    const float* x      = (const float*)d_in[0];
    const int*   ei     = (const int*)d_in[1];
    const float* eattr  = (const float*)d_in[2];
    const float* Wl1    = (const float*)d_in[3];
    const float* Wr1    = (const float*)d_in[4];
    const float* We1    = (const float*)d_in[5];
    const float* att1   = (const float*)d_in[6];
    const float* b1     = (const float*)d_in[7];
    const float* skip_W = (const float*)d_in[8];
    const float* skip_b = (const float*)d_in[9];
    const float* bn_g   = (const float*)d_in[10];
    const float* bn_b   = (const float*)d_in[11];
    const float* Wl2    = (const float*)d_in[12];
    const float* Wr2    = (const float*)d_in[13];
    const float* We2    = (const float*)d_in[14];
    const float* att2   = (const float*)d_in[15];
    const float* b2     = (const float*)d_in[16];
    const float* fc1_W  = (const float*)d_in[17];
    const float* fc1_b  = (const float*)d_in[18];
    const float* Wc     = (const float*)d_in[19];
    const float* bc     = (const float*)d_in[20];
    const float* Wt     = (const float*)d_in[21];
    const float* bt     = (const float*)d_in[22];

    const int N    = in_sizes[0] / 128;
    const int E0   = in_sizes[1] / 2;
    const int Etot = E0 + N;

    // -------- workspace layout (floats) --------
    float* ws = (float*)d_ws;
    size_t o = 0;
    float*    xl1   = ws + o; o += (size_t)N * 128;
    float*    xr1   = ws + o; o += (size_t)N * 128;   // reused for gat2-era buffers
    float*    hacc  = ws + o; o += (size_t)N * 128;
    float*    alpha = ws + o; o += (size_t)Etot * 4;  // alpha/ex (gat1: xH, gat2: first Etot)
    float*    eaf   = ws + o; o += (size_t)Etot;
    float*    deg   = ws + o; o += (size_t)N;
    float*    asum  = ws + o; o += (size_t)N;
    unsigned* amax  = (unsigned*)(ws + o); o += (size_t)N * 4;
    float*    den   = ws + o; o += (size_t)N * 4;
    float*    stats = ws + o; o += 256;
    float*    biasc = ws + o; o += 128;
    (void)ws_size; (void)n_in; (void)out_size;

    // gat2-era buffers overlap the (then-dead) xr1 region
    float* xl2  = xr1;
    float* xr2  = xr1 + (size_t)N * 32;
    float* out2 = xr1 + (size_t)N * 64;
    float* hl   = xr1 + (size_t)N * 96;

    float* outp  = (float*)d_out;              // N x 48
    float* w_out = (float*)d_out + (size_t)N * 48;  // Etot attention weights

    const unsigned ENC_NEG_INF = 0x007FFFFFu;  // enc(-inf)
    const int TPB = 256;
    auto cdiv = [](long long a, long long b) { return (int)((a + b - 1) / b); };

    auto gemm = [&](const float* A, const float* W, const float* bias, float* out,
                    int M, int K, int Nc, int act) {
        int tiles = (M / 16) * (Nc / 16);
        gemm_wmma_f32<<<cdiv(tiles, 8), TPB, 0, stream>>>(A, W, bias, out, M, K, Nc, act);
    };

    // 1) self-loop stats + mean edge attr
    fill_f32<<<cdiv(2LL * N, TPB), TPB, 0, stream>>>(deg, 0.0f, 2LL * N);  // deg+asum contiguous
    edge_stats<<<cdiv(E0, TPB), TPB, 0, stream>>>(ei, eattr, eaf, deg, asum, E0);
    mean_loop<<<cdiv(N, TPB), TPB, 0, stream>>>(deg, asum, eaf + E0, N);

    // 2) node GEMMs for gat1 + fused skip: hacc = x@skip_W + (skip_b + b1)
    add_vec<<<1, 128, 0, stream>>>(b1, skip_b, biasc, 128);
    gemm(x, Wl1, nullptr, xl1, N, 128, 128, 0);
    gemm(x, Wr1, nullptr, xr1, N, 128, 128, 0);
    gemm(x, skip_W, biasc, hacc, N, 128, 128, 0);

    // 3) gat1 segment softmax + scatter into hacc
    fill_u32<<<cdiv(4LL * N, TPB), TPB, 0, stream>>>(amax, ENC_NEG_INF, 4LL * N);
    fill_f32<<<cdiv(4LL * N, TPB), TPB, 0, stream>>>(den, 0.0f, 4LL * N);
    gat1_edge_alpha<<<cdiv((long long)Etot * 32, TPB), TPB, 0, stream>>>(
        ei, eaf, xl1, xr1, We1, att1, alpha, amax, E0, Etot);
    softmax_num<<<cdiv((long long)Etot * 4, TPB), TPB, 0, stream>>>(
        ei + E0, amax, alpha, den, E0, Etot, 4);
    gat1_scatter<<<cdiv((long long)Etot * 32, TPB), TPB, 0, stream>>>(
        ei, xl1, alpha, den, hacc, E0, Etot);

    // 4) batchnorm + ELU (in place on hacc)
    fill_f32<<<1, 256, 0, stream>>>(stats, 0.0f, 256);
    bn_stats<<<1024, 256, 0, stream>>>(hacc, stats, N);
    bn_apply_elu<<<cdiv((long long)N * 128, TPB), TPB, 0, stream>>>(
        hacc, stats, bn_g, bn_b, 1.0f / (float)N, (long long)N * 128);

    // 5) gat2
    gemm(hacc, Wl2, nullptr, xl2, N, 128, 32, 0);
    gemm(hacc, Wr2, nullptr, xr2, N, 128, 32, 0);
    fill_u32<<<cdiv((long long)N, TPB), TPB, 0, stream>>>(amax, ENC_NEG_INF, N);
    fill_f32<<<cdiv((long long)N, TPB), TPB, 0, stream>>>(den, 0.0f, N);
    fill_f32<<<cdiv((long long)N * 32, TPB), TPB, 0, stream>>>(out2, 0.0f, (long long)N * 32);
    gat2_edge_alpha<<<cdiv((long long)Etot * 32, TPB), TPB, 0, stream>>>(
        ei, eaf, xl2, xr2, We2, att2, alpha, amax, E0, Etot);
    softmax_num<<<cdiv((long long)Etot, TPB), TPB, 0, stream>>>(
        ei + E0, amax, alpha, den, E0, Etot, 1);
    gat2_scatter<<<cdiv((long long)Etot * 32, TPB), TPB, 0, stream>>>(
        ei, xl2, alpha, den, out2, w_out, E0, Etot);

    // 6) +b2, ELU, fc1 (+relu), heads with log_softmax
    bias_elu<<<cdiv((long long)N * 32, TPB), TPB, 0, stream>>>(out2, b2, (long long)N * 32, 32);
    gemm(out2, fc1_W, fc1_b, hl, N, 32, 32, 1);
    heads_kernel<<<cdiv(N, TPB), TPB, 0, stream>>>(hl, Wc, bc, Wt, bt, outp, N);
}